// TimeAwareMultiHeadAttention2_39994735460882
// MI455X (gfx1250) — compile-verified
//
#include <hip/hip_runtime.h>
#include <hip/hip_bf16.h>

// ---------------- problem constants ----------------
#define B_   8
#define L_   200
#define H_   128
#define NH_  2
#define HD_  64
#define LP_  208          // L padded to multiple of 16
#define NT_  13           // LP_/16
#define TV_  32           // time vocab

typedef __attribute__((ext_vector_type(2))) float v2f;
typedef __attribute__((ext_vector_type(8))) float v8f;

// fp32 WMMA: D[16x16] = A[16x4] * B[4x16] + C   (full fp32 precision)
__device__ __forceinline__ v8f wmma4(v2f a, v2f b, v8f c) {
  return __builtin_amdgcn_wmma_f32_16x16x4_f32(
      /*neg_a=*/false, a, /*neg_b=*/false, b,
      /*c_mod=*/(short)0, c, /*reuse_a=*/false, /*reuse_b=*/false);
}

// workspace sizes (floats)
#define QS_ (B_ * NH_ * LP_ * HD_)   // 212992
#define SS_ (B_ * NH_ * LP_ * LP_)   // 692224

// ==========================================================================
// Kernel A: q = X@Wq+bq ; kk = X@Wk+bk+kpos ; vv = X@Wv+bv+vpos
// head-split padded layout [B, NH, LP, HD]. grid = B_*NT_, block = 256.
// ==========================================================================
__global__ void qkv_kernel(const float* __restrict__ X,
                           const float* __restrict__ Wq, const float* __restrict__ bq,
                           const float* __restrict__ Wk, const float* __restrict__ bk,
                           const float* __restrict__ Wv, const float* __restrict__ bv,
                           const float* __restrict__ kpt, const float* __restrict__ vpt,
                           float* __restrict__ qb, float* __restrict__ kkb,
                           float* __restrict__ vvb) {
  __shared__ float Xs[16][H_];                 // 8 KB
  const int blk = blockIdx.x;
  const int b = blk / NT_;
  const int qt = blk % NT_;
  const int tid = threadIdx.x;

  for (int i = tid; i < 16 * H_; i += 256) {
    int r = i >> 7, c = i & 127;
    int gr = qt * 16 + r;
    Xs[r][c] = (gr < L_) ? X[(b * L_ + gr) * H_ + c] : 0.0f;
  }
  __syncthreads();

  const int wave = tid >> 5, lane = tid & 31;
  const int n16 = lane & 15, kh = lane >> 4;
  const int n = wave * 16 + n16;               // output column 0..127
  const int hh = n >> 6, d = n & 63;

  const float* Ws[3] = {Wq, Wk, Wv};
  const float* bs[3] = {bq, bk, bv};
  float* dsts[3]     = {qb, kkb, vvb};

  for (int mtx = 0; mtx < 3; ++mtx) {
    const float* W = Ws[mtx];
    v8f acc = {0.f, 0.f, 0.f, 0.f, 0.f, 0.f, 0.f, 0.f};
    for (int kb = 0; kb < H_; kb += 4) {
      int k0 = kb + 2 * kh;
      v2f a  = {Xs[n16][k0], Xs[n16][k0 + 1]};
      v2f bf = {W[k0 * H_ + n], W[(k0 + 1) * H_ + n]};
      acc = wmma4(a, bf, acc);
    }
    float bias = bs[mtx][n];
    float* dst = dsts[mtx];
    for (int i = 0; i < 8; ++i) {
      int m = i + 8 * kh;
      int gr = qt * 16 + m;
      float val = acc[i] + bias;
      if (mtx == 1 && gr < L_) val += kpt[gr * H_ + n];
      if (mtx == 2 && gr < L_) val += vpt[gr * H_ + n];
      dst[((b * NH_ + hh) * LP_ + gr) * HD_ + d] = val;
    }
  }
}

// ==========================================================================
// Kernel B: scores = (q·kk^T + gather(q·ktt^T, diff)) / sqrt(HD) + mask
// grid = B_*NH_*NT_, block = 256 (8 waves).
// ==========================================================================
__global__ void scores_kernel(const float* __restrict__ qb,
                              const float* __restrict__ kkb,
                              const int* __restrict__ tseq,
                              const float* __restrict__ mask,
                              const float* __restrict__ ktt_tab,
                              float* __restrict__ scores) {
  __shared__ float Qs[16][HD_];    // 4 KB
  __shared__ float KTT[TV_][HD_];  // 8 KB
  __shared__ float QT[16][TV_];    // 2 KB
  __shared__ int   TQ[16];
  __shared__ int   TK[LP_];

  const int blk = blockIdx.x;
  const int qt = blk % NT_;
  const int bh = blk / NT_;
  const int h = bh % NH_;
  const int b = bh / NH_;
  const int tid = threadIdx.x;

  const float* qbase  = qb  + (size_t)bh * LP_ * HD_;
  const float* kkbase = kkb + (size_t)bh * LP_ * HD_;

  for (int i = tid; i < 16 * HD_; i += 256) {
    int r = i >> 6, c = i & 63;
    Qs[r][c] = qbase[(qt * 16 + r) * HD_ + c];
  }
  for (int i = tid; i < TV_ * HD_; i += 256) {
    int v = i >> 6, c = i & 63;
    KTT[v][c] = ktt_tab[v * H_ + h * HD_ + c];
  }
  for (int i = tid; i < LP_; i += 256) TK[i] = (i < L_) ? tseq[b * L_ + i] : 0;
  if (tid < 16) {
    int gr = qt * 16 + tid;
    TQ[tid] = (gr < L_) ? tseq[b * L_ + gr] : 0;
  }
  __syncthreads();

  const int wave = tid >> 5, lane = tid & 31;
  const int n16 = lane & 15, kh = lane >> 4;

  // qt[16][32] = Q[16x64] @ KTT^T[64x32] ; waves 0,1 each do one 16-col tile
  if (wave < 2) {
    v8f acc = {0.f, 0.f, 0.f, 0.f, 0.f, 0.f, 0.f, 0.f};
    const int v = wave * 16 + n16;
    for (int kb = 0; kb < HD_; kb += 4) {
      int k0 = kb + 2 * kh;
      v2f a  = {Qs[n16][k0], Qs[n16][k0 + 1]};
      v2f bf = {KTT[v][k0], KTT[v][k0 + 1]};
      acc = wmma4(a, bf, acc);
    }
    for (int i = 0; i < 8; ++i) QT[i + 8 * kh][v] = acc[i];
  }
  __syncthreads();

  float* sbase = scores + (size_t)bh * LP_ * LP_ + (size_t)qt * 16 * LP_;

  for (int kt = wave; kt < NT_; kt += 8) {
    v8f acc = {0.f, 0.f, 0.f, 0.f, 0.f, 0.f, 0.f, 0.f};
    const int kg = kt * 16 + n16;                 // global key column
    for (int kb = 0; kb < HD_; kb += 4) {
      int k0 = kb + 2 * kh;
      v2f a  = {Qs[n16][k0], Qs[n16][k0 + 1]};
      v2f bf = {kkbase[(size_t)kg * HD_ + k0], kkbase[(size_t)kg * HD_ + k0 + 1]};
      acc = wmma4(a, bf, acc);
    }
    for (int i = 0; i < 8; ++i) {
      int m = i + 8 * kh;
      int qg = qt * 16 + m;
      if (kg < L_) {
        int dt = TQ[m] - TK[kg];
        if (dt < 0) dt = -dt;
        int di = (int)log1pf((float)dt);
        if (di > TV_ - 1) di = TV_ - 1;
        float val = (acc[i] + QT[m][di]) * 0.125f;   // 1/sqrt(64)
        if (qg < L_) val += mask[((size_t)b * L_ + qg) * L_ + kg];
        sbase[(size_t)m * LP_ + kg] = val;
      } else {
        sbase[(size_t)m * LP_ + kg] = 0.0f;          // padded col (never read)
      }
    }
  }
}

// ==========================================================================
// Kernel C: softmax over keys. grid = B_*NH_*LP_ (one row each), block = 256.
// Padded rows/cols -> 0 so downstream WMMA tiles stay clean.
// ==========================================================================
__global__ void softmax_kernel(const float* __restrict__ scores,
                               float* __restrict__ probs) {
  __shared__ float red[256];
  const int row = blockIdx.x;          // = bh*LP_ + q  (matches layout)
  const int q = row % LP_;
  const int tid = threadIdx.x;
  const float* srow = scores + (size_t)row * LP_;
  float* prow = probs + (size_t)row * LP_;

  if (q >= L_) {
    for (int k = tid; k < LP_; k += 256) prow[k] = 0.0f;
    return;
  }

  float v = (tid < L_) ? srow[tid] : -3.0e38f;
  red[tid] = v;
  __syncthreads();
  for (int s = 128; s > 0; s >>= 1) {
    if (tid < s) red[tid] = fmaxf(red[tid], red[tid + s]);
    __syncthreads();
  }
  float mx = red[0];
  __syncthreads();

  float e = (tid < L_) ? expf(v - mx) : 0.0f;
  red[tid] = e;
  __syncthreads();
  for (int s = 128; s > 0; s >>= 1) {
    if (tid < s) red[tid] += red[tid + s];
    __syncthreads();
  }
  float inv = 1.0f / red[0];
  if (tid < LP_) prow[tid] = (tid < L_) ? e * inv : 0.0f;
}

// ==========================================================================
// Kernel D: ctx = probs @ vv  +  bucket(scores by diff) @ vtt
// grid = B_*NH_*NT_, block = 256. Waves 0-3: GEMM N-tiles; waves 4-7: buckets.
// ==========================================================================
__global__ void ctx_kernel(const float* __restrict__ probs,
                           const float* __restrict__ scores,
                           const float* __restrict__ vvb,
                           const int* __restrict__ tseq,
                           const float* __restrict__ vtt_tab,
                           float* __restrict__ ctx) {
  __shared__ float SB[16][TV_];    // 2 KB bucket sums
  __shared__ int   TQ[16];
  __shared__ int   TK[LP_];

  const int blk = blockIdx.x;
  const int qt = blk % NT_;
  const int bh = blk / NT_;
  const int h = bh % NH_;
  const int b = bh / NH_;
  const int tid = threadIdx.x;

  const float* prow = probs  + (size_t)bh * LP_ * LP_ + (size_t)qt * 16 * LP_;
  const float* srow = scores + (size_t)bh * LP_ * LP_ + (size_t)qt * 16 * LP_;
  const float* vv   = vvb + (size_t)bh * LP_ * HD_;
  float* cbase      = ctx + ((size_t)bh * LP_ + qt * 16) * HD_;

  for (int i = tid; i < LP_; i += 256) TK[i] = (i < L_) ? tseq[b * L_ + i] : 0;
  if (tid < 16) {
    int gr = qt * 16 + tid;
    TQ[tid] = (gr < L_) ? tseq[b * L_ + gr] : 0;
  }
  __syncthreads();

  const int wave = tid >> 5, lane = tid & 31;
  const int n16 = lane & 15, kh = lane >> 4;

  v8f acc = {0.f, 0.f, 0.f, 0.f, 0.f, 0.f, 0.f, 0.f};

  if (wave < 4) {
    // main GEMM: D[16 x 16cols] over K = LP_
    const int n = wave * 16 + n16;
    for (int kt = 0; kt < NT_; ++kt) {
      for (int kb = 0; kb < 16; kb += 4) {
        int k0 = kt * 16 + kb + 2 * kh;
        v2f a  = {prow[(size_t)n16 * LP_ + k0], prow[(size_t)n16 * LP_ + k0 + 1]};
        v2f bf = {vv[(size_t)k0 * HD_ + n], vv[(size_t)(k0 + 1) * HD_ + n]};
        acc = wmma4(a, bf, acc);
      }
    }
  } else {
    // bucket sums: rows 4*(wave-4)..+3, lane owns bucket index `lane`
    for (int r = (wave - 4) * 4; r < (wave - 4) * 4 + 4; ++r) {
      float s = 0.0f;
      const int tq = TQ[r];
      for (int k = 0; k < L_; ++k) {
        int dt = tq - TK[k];
        if (dt < 0) dt = -dt;
        int di = (int)log1pf((float)dt);
        if (di == lane) s += srow[(size_t)r * LP_ + k];
      }
      SB[r][lane] = s;
    }
  }
  __syncthreads();

  if (wave < 4) {
    const int n = wave * 16 + n16;               // d index 0..63
    const float* vtt = vtt_tab + h * HD_;        // vtt[v][d] = vtt[v*H_ + d]
    for (int kb = 0; kb < TV_; kb += 4) {
      int k0 = kb + 2 * kh;
      v2f a  = {SB[n16][k0], SB[n16][k0 + 1]};
      v2f bf = {vtt[(size_t)k0 * H_ + n], vtt[(size_t)(k0 + 1) * H_ + n]};
      acc = wmma4(a, bf, acc);
    }
    for (int i = 0; i < 8; ++i) {
      int m = i + 8 * kh;
      cbase[(size_t)m * HD_ + n] = acc[i];
    }
  }
}

// ==========================================================================
// Kernel E: hidden = merge_heads(ctx) @ Wd + bd + X ; LayerNorm -> out
// grid = (B_*L_)/16 = 100 exactly, block = 256.
// ==========================================================================
__global__ void out_kernel(const float* __restrict__ ctx,
                           const float* __restrict__ Wd, const float* __restrict__ dbias,
                           const float* __restrict__ X,
                           const float* __restrict__ ln_g, const float* __restrict__ ln_b,
                           float* __restrict__ out) {
  __shared__ float HID[16][H_];   // 8 KB
  const int tile = blockIdx.x;
  const int tid = threadIdx.x;
  const int wave = tid >> 5, lane = tid & 31;
  const int n16 = lane & 15, kh = lane >> 4;
  const int n = wave * 16 + n16;

  // A row rg = tile*16 + n16 ; k -> (head = k>>6, d = k&63)
  const int rg_a = tile * 16 + n16;          // < 1600 always (100*16 == B_*L_)
  const int bb = rg_a / L_, ll = rg_a % L_;

  v8f acc = {0.f, 0.f, 0.f, 0.f, 0.f, 0.f, 0.f, 0.f};
  for (int kb = 0; kb < H_; kb += 4) {
    int k0 = kb + 2 * kh;
    v2f a = {ctx[(((size_t)bb * NH_ + (k0 >> 6)) * LP_ + ll) * HD_ + (k0 & 63)],
             ctx[(((size_t)bb * NH_ + ((k0 + 1) >> 6)) * LP_ + ll) * HD_ + ((k0 + 1) & 63)]};
    v2f bf = {Wd[(size_t)k0 * H_ + n], Wd[(size_t)(k0 + 1) * H_ + n]};
    acc = wmma4(a, bf, acc);
  }
  const float bias = dbias[n];
  for (int i = 0; i < 8; ++i) {
    int m = i + 8 * kh;
    int rg = tile * 16 + m;
    HID[m][n] = acc[i] + bias + X[(size_t)rg * H_ + n];
  }
  __syncthreads();

  if (tid < 16) {
    float mu = 0.0f;
    for (int c = 0; c < H_; ++c) mu += HID[tid][c];
    mu *= (1.0f / H_);
    float var = 0.0f;
    for (int c = 0; c < H_; ++c) {
      float d = HID[tid][c] - mu;
      var += d * d;
    }
    var *= (1.0f / H_);
    float inv = rsqrtf(var + 1e-12f);
    int rg = tile * 16 + tid;
    for (int c = 0; c < H_; ++c)
      out[(size_t)rg * H_ + c] = (HID[tid][c] - mu) * inv * ln_g[c] + ln_b[c];
  }
}

// ==========================================================================
extern "C" void kernel_launch(void* const* d_in, const int* in_sizes, int n_in,
                              void* d_out, int out_size, void* d_ws, size_t ws_size,
                              hipStream_t stream) {
  const float* X    = (const float*)d_in[0];
  const int*   tseq = (const int*)d_in[1];
  const float* mask = (const float*)d_in[2];
  const float* Wq = (const float*)d_in[3];
  const float* bq = (const float*)d_in[4];
  const float* Wk = (const float*)d_in[5];
  const float* bk = (const float*)d_in[6];
  const float* Wv = (const float*)d_in[7];
  const float* bv = (const float*)d_in[8];
  const float* Wd = (const float*)d_in[9];
  const float* bd = (const float*)d_in[10];
  const float* ln_g = (const float*)d_in[11];
  const float* ln_b = (const float*)d_in[12];
  const float* ktt  = (const float*)d_in[13];
  const float* vtt  = (const float*)d_in[14];
  const float* kpt  = (const float*)d_in[15];
  const float* vpt  = (const float*)d_in[16];
  float* out = (float*)d_out;

  float* ws     = (float*)d_ws;
  float* qb     = ws;                 // [B,NH,LP,HD]
  float* kkb    = qb + QS_;           // k + kpos
  float* vvb    = kkb + QS_;          // v + vpos
  float* scores = vvb + QS_;          // [B*NH, LP, LP]
  float* probs  = scores + SS_;
  float* ctx    = probs + SS_;        // [B,NH,LP,HD]

  qkv_kernel<<<B_ * NT_, 256, 0, stream>>>(X, Wq, bq, Wk, bk, Wv, bv, kpt, vpt,
                                           qb, kkb, vvb);
  scores_kernel<<<B_ * NH_ * NT_, 256, 0, stream>>>(qb, kkb, tseq, mask, ktt, scores);
  softmax_kernel<<<B_ * NH_ * LP_, 256, 0, stream>>>(scores, probs);
  ctx_kernel<<<B_ * NH_ * NT_, 256, 0, stream>>>(probs, scores, vvb, tseq, vtt, ctx);
  out_kernel<<<(B_ * L_) / 16, 256, 0, stream>>>(ctx, Wd, bd, X, ln_g, ln_b, out);
}